// CAM_Module_85985245265972
// MI455X (gfx1250) — compile-verified
//
#include <hip/hip_runtime.h>

// Problem constants from the reference: x is (2, B, C, G, H, W) fp32.
constexpr int B = 8;
constexpr int C = 512;
constexpr int N = 4096;   // G*H*W = 4*32*32

typedef float v2f __attribute__((ext_vector_type(2)));
typedef float v8f __attribute__((ext_vector_type(8)));

// Builtin-compatible vector-of-4-int with explicit address spaces.
typedef int v4i __attribute__((__vector_size__(16)));
typedef __attribute__((address_space(1))) v4i* gv4i_p;   // global (AS1)
typedef __attribute__((address_space(3))) v4i* lv4i_p;   // LDS (AS3)

static __device__ __forceinline__ v8f wmma_f32(v2f a, v2f b, v8f c) {
    // V_WMMA_F32_16X16X4_F32 : D = A(16x4) * B(4x16) + C(16x16), fp32 throughout.
    return __builtin_amdgcn_wmma_f32_16x16x4_f32(false, a, false, b, (short)0, c,
                                                 false, false);
}

// Async global->LDS 128-bit copy (GLOBAL_LOAD_ASYNC_TO_LDS_B128, ASYNCcnt).
static __device__ __forceinline__ void async_cp16(void* lds, const void* g) {
    __builtin_amdgcn_global_load_async_to_lds_b128((gv4i_p)g, (lv4i_p)lds,
                                                   /*offset=*/0, /*cpol=*/0);
}

static __device__ __forceinline__ void wait_async_all() {
    __builtin_amdgcn_s_wait_asynccnt(0);
}

// ---------------------------------------------------------------------------
// Stage 1: complex Gram matrices
//   er[b,c,d] = sum_n qr*qr - qi*qi     ei[b,c,d] = sum_n qr*qi + qi*qr
// grid (C/64, C/64, B), block 256 (8 waves). 64x64 output tile per block.
// f32 WMMA has no A/B negate, so accumulate P=ArBr, Q=AiBi, S=ArBi+AiBr.
// Double-buffered LDS fed by async B128 copies.
// ---------------------------------------------------------------------------
__global__ __launch_bounds__(256) void gram_kernel(const float* __restrict__ x,
                                                   float* __restrict__ er,
                                                   float* __restrict__ ei) {
    // Row stride 36 floats: the 16-lane b64 column-fragment reads touch all
    // 64 banks exactly once -> conflict free.
    __shared__ float sMr[2][64][36];
    __shared__ float sMi[2][64][36];
    __shared__ float sNr[2][64][36];
    __shared__ float sNi[2][64][36];

    const int tid = threadIdx.x;
    const int n0  = blockIdx.x * 64;
    const int m0  = blockIdx.y * 64;
    const int b   = blockIdx.z;

    const float* qr = x + (size_t)b * C * N;            // x[0][b]
    const float* qi = x + (size_t)(B + b) * C * N;      // x[1][b]

    const int w        = tid >> 5;
    const int lane     = tid & 31;
    const int rowStrip = (w & 3) * 16;    // which 16-row strip of the 64
    const int colHalf  = (w >> 2) * 32;   // which 32-col half of the 64
    const int rl       = lane & 15;
    const int kh       = (lane >> 4) * 2; // K offset 0 / 2 per ISA A/B layout

    auto issueTiles = [&](int buf, int k0) {
#pragma unroll
        for (int rep = 0; rep < 2; ++rep) {
            const int idx = tid + rep * 256;       // 0..511
            const int row = idx >> 3;              // 8 float4 per 32-col row
            const int c4  = (idx & 7) << 2;
            const size_t offM = (size_t)(m0 + row) * N + k0 + c4;
            const size_t offN = (size_t)(n0 + row) * N + k0 + c4;
            async_cp16(&sMr[buf][row][c4], qr + offM);
            async_cp16(&sMi[buf][row][c4], qi + offM);
            async_cp16(&sNr[buf][row][c4], qr + offN);
            async_cp16(&sNi[buf][row][c4], qi + offN);
        }
    };

    const v8f zero8 = {0.f, 0.f, 0.f, 0.f, 0.f, 0.f, 0.f, 0.f};
    v8f accP[2] = {zero8, zero8};
    v8f accQ[2] = {zero8, zero8};
    v8f accS[2] = {zero8, zero8};

    issueTiles(0, 0);

    for (int k0 = 0; k0 < N; k0 += 32) {
        const int cur = (k0 >> 5) & 1;
        wait_async_all();      // my async copies for buf[cur] have landed
        __syncthreads();       // all waves' copies visible; prev buffer free
        if (k0 + 32 < N) issueTiles(cur ^ 1, k0 + 32);

#pragma unroll
        for (int kk = 0; kk < 8; ++kk) {
            const int k4 = kk * 4 + kh;
            const int ml = rowStrip + rl;
            const v2f ar = *(const v2f*)&sMr[cur][ml][k4];
            const v2f ai = *(const v2f*)&sMi[cur][ml][k4];
#pragma unroll
            for (int t = 0; t < 2; ++t) {
                const int nl = colHalf + t * 16 + rl;
                const v2f br = *(const v2f*)&sNr[cur][nl][k4];
                const v2f bi = *(const v2f*)&sNi[cur][nl][k4];
                accP[t] = wmma_f32(ar, br, accP[t]);
                accQ[t] = wmma_f32(ai, bi, accQ[t]);
                accS[t] = wmma_f32(ar, bi, accS[t]);
                accS[t] = wmma_f32(ai, br, accS[t]);
            }
        }
    }

    // C/D layout: vgpr j, lanes 0-15 -> M=j, lanes 16-31 -> M=j+8; N=lane%16.
    const int rhalf = (lane >> 4) * 8;
#pragma unroll
    for (int t = 0; t < 2; ++t) {
        const int gn = n0 + colHalf + t * 16 + rl;
#pragma unroll
        for (int j = 0; j < 8; ++j) {
            const int gm = m0 + rowStrip + rhalf + j;
            const size_t o = ((size_t)b * C + gm) * C + gn;
            er[o] = accP[t][j] - accQ[t][j];
            ei[o] = accS[t][j];
        }
    }
}

// ---------------------------------------------------------------------------
// Stage 2: attention = softmax( (max(er)-er)^2 + (max(ei)-ei)^2 ) per row.
// One 256-thread block per (b,c) row of length C=512. Writes att over er.
// ---------------------------------------------------------------------------
__global__ __launch_bounds__(256) void softmax_kernel(float* __restrict__ er,
                                                      const float* __restrict__ ei) {
    __shared__ float red[256];
    const int tid = threadIdx.x;
    const size_t row = blockIdx.x;            // b*C + c
    float*       rr = er + row * C;
    const float* ri = ei + row * C;

    const float r0 = rr[tid], r1 = rr[tid + 256];
    const float i0 = ri[tid], i1 = ri[tid + 256];

    auto bmax = [&](float v) -> float {
        red[tid] = v; __syncthreads();
        for (int s = 128; s > 0; s >>= 1) {
            if (tid < s) red[tid] = fmaxf(red[tid], red[tid + s]);
            __syncthreads();
        }
        const float r = red[0]; __syncthreads();
        return r;
    };
    auto bsum = [&](float v) -> float {
        red[tid] = v; __syncthreads();
        for (int s = 128; s > 0; s >>= 1) {
            if (tid < s) red[tid] += red[tid + s];
            __syncthreads();
        }
        const float r = red[0]; __syncthreads();
        return r;
    };

    const float maxr = bmax(fmaxf(r0, r1));
    const float maxi = bmax(fmaxf(i0, i1));

    const float a0 = maxr - r0, c0 = maxi - i0;
    const float a1 = maxr - r1, c1 = maxi - i1;
    const float f0 = a0 * a0 + c0 * c0;
    const float f1 = a1 * a1 + c1 * c1;

    const float M  = bmax(fmaxf(f0, f1));
    const float e0 = __expf(f0 - M);
    const float e1 = __expf(f1 - M);
    const float S  = bsum(e0 + e1);
    const float inv = 1.0f / S;

    rr[tid]       = e0 * inv;
    rr[tid + 256] = e1 * inv;
}

// ---------------------------------------------------------------------------
// Stage 3: out = gamma * (att @ q) + x for both complex components.
// grid (N/64, C/64, B), block 256. 64(rows c) x 64(cols n) tile per block.
// Double-buffered LDS fed by async B128 copies.
// ---------------------------------------------------------------------------
__global__ __launch_bounds__(256) void out_kernel(const float* __restrict__ x,
                                                  const float* __restrict__ att,
                                                  const float* __restrict__ gamma,
                                                  float* __restrict__ out) {
    __shared__ float sA[2][64][36];    // att tile [m][k]
    __shared__ float sQr[2][32][72];   // q tiles [k][n]; stride 72 keeps the
    __shared__ float sQi[2][32][72];   // two 16-lane row reads on disjoint banks

    const int tid = threadIdx.x;
    const int n0  = blockIdx.x * 64;
    const int c0  = blockIdx.y * 64;
    const int b   = blockIdx.z;

    const float* qr = x + (size_t)b * C * N;
    const float* qi = x + (size_t)(B + b) * C * N;
    const float* A  = att + (size_t)b * C * C;

    const int w        = tid >> 5;
    const int lane     = tid & 31;
    const int rowStrip = (w & 3) * 16;
    const int colHalf  = (w >> 2) * 32;
    const int rl       = lane & 15;
    const int kh       = (lane >> 4) * 2;

    auto issueTiles = [&](int buf, int k0) {
#pragma unroll
        for (int rep = 0; rep < 2; ++rep) {
            const int idx = tid + rep * 256;
            // att tile: 64 rows x 32 cols
            const int arow = idx >> 3;
            const int ac4  = (idx & 7) << 2;
            async_cp16(&sA[buf][arow][ac4], A + (size_t)(c0 + arow) * C + k0 + ac4);
            // q tiles: 32 rows x 64 cols
            const int kk = idx >> 4;
            const int n4 = (idx & 15) << 2;
            const size_t off = (size_t)(k0 + kk) * N + n0 + n4;
            async_cp16(&sQr[buf][kk][n4], qr + off);
            async_cp16(&sQi[buf][kk][n4], qi + off);
        }
    };

    const v8f zero8 = {0.f, 0.f, 0.f, 0.f, 0.f, 0.f, 0.f, 0.f};
    v8f accR[2] = {zero8, zero8};
    v8f accI[2] = {zero8, zero8};

    issueTiles(0, 0);

    for (int k0 = 0; k0 < C; k0 += 32) {
        const int cur = (k0 >> 5) & 1;
        wait_async_all();
        __syncthreads();
        if (k0 + 32 < C) issueTiles(cur ^ 1, k0 + 32);

#pragma unroll
        for (int kk = 0; kk < 8; ++kk) {
            const int k4 = kk * 4 + kh;
            const v2f aa = *(const v2f*)&sA[cur][rowStrip + rl][k4];
#pragma unroll
            for (int t = 0; t < 2; ++t) {
                const int nl = colHalf + t * 16 + rl;
                v2f br; br.x = sQr[cur][k4][nl]; br.y = sQr[cur][k4 + 1][nl];
                v2f bi; bi.x = sQi[cur][k4][nl]; bi.y = sQi[cur][k4 + 1][nl];
                accR[t] = wmma_f32(aa, br, accR[t]);
                accI[t] = wmma_f32(aa, bi, accI[t]);
            }
        }
    }

    const float g = gamma[0];
    const int rhalf = (lane >> 4) * 8;
#pragma unroll
    for (int t = 0; t < 2; ++t) {
        const int gn = n0 + colHalf + t * 16 + rl;
#pragma unroll
        for (int j = 0; j < 8; ++j) {
            const int gm = c0 + rowStrip + rhalf + j;
            const size_t oo = (size_t)gm * N + gn;        // within (C,N) slab
            out[(size_t)b * C * N + oo]       = g * accR[t][j] + qr[oo];
            out[(size_t)(B + b) * C * N + oo] = g * accI[t][j] + qi[oo];
        }
    }
}

// ---------------------------------------------------------------------------
extern "C" void kernel_launch(void* const* d_in, const int* in_sizes, int n_in,
                              void* d_out, int out_size, void* d_ws, size_t ws_size,
                              hipStream_t stream) {
    (void)in_sizes; (void)n_in; (void)out_size; (void)ws_size;
    const float* x     = (const float*)d_in[0];
    const float* gamma = (const float*)d_in[1];
    float* out = (float*)d_out;

    float* er = (float*)d_ws;                     // B*C*C floats (8 MB)
    float* ei = er + (size_t)B * C * C;           // B*C*C floats (8 MB)

    const dim3 blk(256);
    gram_kernel<<<dim3(C / 64, C / 64, B), blk, 0, stream>>>(x, er, ei);
    softmax_kernel<<<dim3(B * C), blk, 0, stream>>>(er, ei);
    out_kernel<<<dim3(N / 64, C / 64, B), blk, 0, stream>>>(x, er, gamma, out);
}